// GATNet_21912923144584
// MI455X (gfx1250) — compile-verified
//
#include <hip/hip_runtime.h>
#include <hip/hip_bf16.h>
#include <cstdint>

typedef __attribute__((ext_vector_type(2))) float v2f;
typedef __attribute__((ext_vector_type(8))) float v8f;

#define NEG_SLOPE 0.2f

// ---- monotonic float<->u32 key for atomicMax-based segment max ----
__device__ __forceinline__ unsigned enc_key(float f) {
  unsigned i = __float_as_uint(f);
  return (i & 0x80000000u) ? ~i : (i | 0x80000000u);
}
__device__ __forceinline__ float dec_key(unsigned k) {
  unsigned i = (k & 0x80000000u) ? (k & 0x7fffffffu) : ~k;
  float f = __uint_as_float(i);
  // reference: where(isfinite(m), m, 0) ; key==0 decodes to -NaN -> 0
  if (!(f >= -__FLT_MAX__ && f <= __FLT_MAX__)) f = 0.0f;
  return f;
}

__device__ __forceinline__ void edge_sd(const int* __restrict__ ei, int e, int E,
                                        int& s, int& d) {
  if (e < E) { s = ei[e]; d = ei[E + e]; }   // edge_index[0,:], edge_index[1,:]
  else       { s = e - E; d = s; }           // appended self loops
}

// ============ WMMA fp32 GEMM: C[M,Nc] = A[M,K] @ B[K,Nc], Nc = NT*16 =========
// Whole B staged in LDS transposed (lds[n*K+k]) -> B K-pair is one ds b64/2addr.
// One wave owns a 16-row panel and register-blocks all NT column tiles.
// Per k-step: load A fragment (global b64) + ALL NT B fragments into distinct
// registers first (so DS loads overlap), then issue NT back-to-back WMMAs.
template <int NT>
__global__ __launch_bounds__(128)
void gat_gemm_wmma(const float* __restrict__ A, const float* __restrict__ B,
                   float* __restrict__ C, int M, int K, int tilesM) {
  extern __shared__ float ldsB[];            // [Nc][K] = B transposed
  const int Nc = NT * 16;

  // cooperative fill: coalesced global read, transposed LDS write
  for (int idx = threadIdx.x; idx < K * Nc; idx += 128) {
    int k = idx / Nc, n = idx - k * Nc;      // B[k][n]
    ldsB[n * K + k] = B[idx];
  }
  __syncthreads();

  int wave = threadIdx.x >> 5;
  int lane = threadIdx.x & 31;
  int tm = blockIdx.x * 4 + wave;
  if (tm >= tilesM) return;                  // wave-uniform: EXEC all-ones below
  int row0 = tm << 4;
  int m = lane & 15;                         // A row in tile / B,C column
  int g = lane >> 4;                         // lane group: K pair / M half

  int arow = row0 + m; if (arow >= M) arow = M - 1;   // clamp, no divergence
  const float* __restrict__ Arow = A + (size_t)arow * K;
  const float* __restrict__ Bcol = ldsB + (size_t)m * K;   // this lane's column

  v8f acc[NT];
#pragma unroll
  for (int nt = 0; nt < NT; ++nt) acc[nt] = (v8f){0.f,0.f,0.f,0.f,0.f,0.f,0.f,0.f};

  for (int k0 = 0; k0 < K; k0 += 4) {
    int ka = k0 + (g << 1);                  // this lane-group's K pair
    v2f a = *(const v2f*)(Arow + ka);        // A(m,ka), A(m,ka+1): global b64
    v2f bfrag[NT];
#pragma unroll
    for (int nt = 0; nt < NT; ++nt)          // distinct regs -> DS loads overlap
      bfrag[nt] = *(const v2f*)(Bcol + (size_t)(nt << 4) * K + ka);
#pragma unroll
    for (int nt = 0; nt < NT; ++nt)
      acc[nt] = __builtin_amdgcn_wmma_f32_16x16x4_f32(false, a, false, bfrag[nt],
                                                      (short)0, acc[nt],
                                                      false, false);
  }

#pragma unroll
  for (int nt = 0; nt < NT; ++nt) {
#pragma unroll
    for (int v = 0; v < 8; ++v) {            // C layout: VGPR v -> M=v (+8, g=1)
      int row = row0 + v + (g << 3);
      if (row < M) C[(size_t)row * Nc + (nt << 4) + m] = acc[nt][v];
    }
  }
}

// ================= per-node attention scores: a = einsum(nhc,hc->nh) ==========
__global__ void gat_att_scores(const float* __restrict__ h,
                               const float* __restrict__ attS,
                               const float* __restrict__ attD,
                               float* __restrict__ aS, float* __restrict__ aD,
                               int N, int H, int Cc) {
  int t = blockIdx.x * blockDim.x + threadIdx.x;
  if (t >= N * H) return;
  int n = t / H, hh = t - n * H;
  const float* __restrict__ hp = h + (size_t)n * H * Cc + (size_t)hh * Cc;
  const float* __restrict__ sp = attS + hh * Cc;
  const float* __restrict__ dp = attD + hh * Cc;
  float ss = 0.f, sd = 0.f;
  for (int c = 0; c < Cc; ++c) { float v = hp[c]; ss += v * sp[c]; sd += v * dp[c]; }
  aS[t] = ss; aD[t] = sd;
}

// ================= edge pass A: segment max of leaky_relu(e) ==================
__global__ void gat_edge_max(const int* __restrict__ ei, int E, int ET,
                             const float* __restrict__ aS, const float* __restrict__ aD,
                             unsigned* __restrict__ mk, int H) {
  int t = blockIdx.x * blockDim.x + threadIdx.x;
  if (t >= ET * H) return;
  int e = t / H, hh = t - e * H;
  int s, d; edge_sd(ei, e, E, s, d);
  float v = aS[s * H + hh] + aD[d * H + hh];
  v = v > 0.f ? v : NEG_SLOPE * v;
  atomicMax(&mk[d * H + hh], enc_key(v));
}

// ================= edge pass B: denom = segment_sum(exp(e - m)) ===============
__global__ void gat_edge_den(const int* __restrict__ ei, int E, int ET,
                             const float* __restrict__ aS, const float* __restrict__ aD,
                             const unsigned* __restrict__ mk, float* __restrict__ den,
                             int H) {
  int t = blockIdx.x * blockDim.x + threadIdx.x;
  if (t >= ET * H) return;
  int e = t / H, hh = t - e * H;
  int s, d; edge_sd(ei, e, E, s, d);
  float v = aS[s * H + hh] + aD[d * H + hh];
  v = v > 0.f ? v : NEG_SLOPE * v;
  int di = d * H + hh;
  atomicAdd(&den[di], __expf(v - dec_key(mk[di])));
}

// ========= edge pass C: out[dst] += alpha * h[src]  (16 floats / thread) ======
__global__ void gat_edge_agg(const int* __restrict__ ei, int E, int ET,
                             const float* __restrict__ aS, const float* __restrict__ aD,
                             const unsigned* __restrict__ mk, const float* __restrict__ den,
                             const float* __restrict__ hsrc, float* __restrict__ outA,
                             int H, int Cc, int chunks) {
  int t = blockIdx.x * blockDim.x + threadIdx.x;
  int hc = H * chunks;
  if (t >= ET * hc) return;
  int e = t / hc, r = t - e * hc;
  int hh = r / chunks, ch = r - hh * chunks;
  int s, d; edge_sd(ei, e, E, s, d);
  float v = aS[s * H + hh] + aD[d * H + hh];
  v = v > 0.f ? v : NEG_SLOPE * v;
  int di = d * H + hh;
  float alpha = __expf(v - dec_key(mk[di])) / den[di];
  const float4* __restrict__ hp =
      (const float4*)(hsrc + (size_t)s * H * Cc + (size_t)hh * Cc + (ch << 4));
  float* __restrict__ op = outA + (size_t)d * H * Cc + (size_t)hh * Cc + (ch << 4);
#pragma unroll
  for (int q = 0; q < 4; ++q) {
    float4 hv = hp[q];
    atomicAdd(op + 4 * q + 0, alpha * hv.x);
    atomicAdd(op + 4 * q + 1, alpha * hv.y);
    atomicAdd(op + 4 * q + 2, alpha * hv.z);
    atomicAdd(op + 4 * q + 3, alpha * hv.w);
  }
}

// ================= epilogues ==================================================
__global__ void gat_bias_elu(float* __restrict__ x, const float* __restrict__ b,
                             int total, int F) {
  int t = blockIdx.x * blockDim.x + threadIdx.x;
  if (t >= total) return;
  float v = x[t] + b[t % F];
  x[t] = v > 0.f ? v : (__expf(v) - 1.f);
}

__global__ void gat_bias_add(float* __restrict__ x, const float* __restrict__ b,
                             int total, int F) {
  int t = blockIdx.x * blockDim.x + threadIdx.x;
  if (t >= total) return;
  x[t] += b[t % F];
}

// ================= host-side orchestration ====================================
extern "C" void kernel_launch(void* const* d_in, const int* in_sizes, int n_in,
                              void* d_out, int out_size, void* d_ws, size_t ws_size,
                              hipStream_t stream) {
  const float* x    = (const float*)d_in[0];
  const int*   ei   = (const int*)d_in[1];
  const float* W1   = (const float*)d_in[2];
  const float* asr1 = (const float*)d_in[3];
  const float* ads1 = (const float*)d_in[4];
  const float* b1   = (const float*)d_in[5];
  const float* W2   = (const float*)d_in[6];
  const float* asr2 = (const float*)d_in[7];
  const float* ads2 = (const float*)d_in[8];
  const float* b2   = (const float*)d_in[9];

  const int N  = in_sizes[0] / 128;
  const int E  = in_sizes[1] / 2;
  const int ET = E + N;                    // with self loops

  // workspace layout (floats)
  float* ws = (float*)d_ws;
  size_t off = 0;
  float* h1   = ws + off; off += (size_t)N * 128;   // x @ W1
  float* agg1 = ws + off; off += (size_t)N * 128;   // layer-1 aggregate -> h1post
  float* h2   = ws + off; off += (size_t)N * 64;    // h1post @ W2
  float* as1  = ws + off; off += (size_t)N * 8;
  float* ad1  = ws + off; off += (size_t)N * 8;
  float* as2  = ws + off; off += (size_t)N;
  float* ad2  = ws + off; off += (size_t)N;
  float* zblk = ws + off;                           // contiguous zero-init block:
  unsigned* mk1 = (unsigned*)(ws + off); off += (size_t)N * 8;
  float*    den1 =           ws + off;   off += (size_t)N * 8;
  unsigned* mk2 = (unsigned*)(ws + off); off += (size_t)N;
  float*    den2 =           ws + off;   off += (size_t)N;

  hipMemsetAsync(agg1, 0, (size_t)N * 128 * sizeof(float), stream);
  hipMemsetAsync(zblk, 0, (size_t)N * 18 * sizeof(float), stream);
  hipMemsetAsync(d_out, 0, (size_t)out_size * sizeof(float), stream);

  const int TB = 256;
  const int K = 128;

  // ---- layer 1 ----
  {
    int tilesM = (N + 15) / 16;                       // wave per 16-row panel
    int blocks = (tilesM + 3) / 4;
    size_t lds = (size_t)K * 128 * sizeof(float);     // W1 transposed: 64 KB
    gat_gemm_wmma<8><<<blocks, 128, lds, stream>>>(x, W1, h1, N, K, tilesM);
  }
  gat_att_scores<<<(N * 8 + TB - 1) / TB, TB, 0, stream>>>(h1, asr1, ads1,
                                                           as1, ad1, N, 8, 16);
  {
    int blocks = (ET * 8 + TB - 1) / TB;
    gat_edge_max<<<blocks, TB, 0, stream>>>(ei, E, ET, as1, ad1, mk1, 8);
    gat_edge_den<<<blocks, TB, 0, stream>>>(ei, E, ET, as1, ad1, mk1, den1, 8);
    gat_edge_agg<<<blocks, TB, 0, stream>>>(ei, E, ET, as1, ad1, mk1, den1,
                                            h1, agg1, 8, 16, 1);
  }
  gat_bias_elu<<<(N * 128 + TB - 1) / TB, TB, 0, stream>>>(agg1, b1, N * 128, 128);

  // ---- layer 2 ----
  {
    int tilesM = (N + 15) / 16;
    int blocks = (tilesM + 3) / 4;
    size_t lds = (size_t)K * 64 * sizeof(float);      // W2 transposed: 32 KB
    gat_gemm_wmma<4><<<blocks, 128, lds, stream>>>(agg1, W2, h2, N, K, tilesM);
  }
  gat_att_scores<<<(N + TB - 1) / TB, TB, 0, stream>>>(h2, asr2, ads2,
                                                       as2, ad2, N, 1, 64);
  {
    int blocks = (ET + TB - 1) / TB;
    gat_edge_max<<<blocks, TB, 0, stream>>>(ei, E, ET, as2, ad2, mk2, 1);
    gat_edge_den<<<blocks, TB, 0, stream>>>(ei, E, ET, as2, ad2, mk2, den2, 1);
    int blocks4 = (ET * 4 + TB - 1) / TB;             // 4 chunks of 16 per edge
    gat_edge_agg<<<blocks4, TB, 0, stream>>>(ei, E, ET, as2, ad2, mk2, den2,
                                             h2, (float*)d_out, 1, 64, 4);
  }
  gat_bias_add<<<(N * 64 + TB - 1) / TB, TB, 0, stream>>>((float*)d_out, b2,
                                                          N * 64, 64);
}